// LogPolar_47038481826470
// MI455X (gfx1250) — compile-verified
//
#include <hip/hip_runtime.h>
#include <math.h>

// Problem constants (match reference)
#define OUT_HW   (512 * 512)     // one image / one output map
#define N_IMG    96              // 32 batches * 3 channels
#define PPT      4               // pixels (r-bins) per thread
#define IMG_GRP  4               // image groups (grid.y)
#define IMG_PER_GRP (N_IMG / IMG_GRP)   // 24

// native clang vector type: accepted by __builtin_nontemporal_store and
// lowers to one 16B global_store_b128
typedef float v4f __attribute__((ext_vector_type(4)));

__global__ __launch_bounds__(256)
void logpolar_remap_kernel(const float* __restrict__ in, float* __restrict__ out)
{
    // ---- which output pixels does this thread own? (4 consecutive r bins) ----
    const int tid   = blockIdx.x * 256 + threadIdx.x;
    const int pix0  = tid * PPT;            // r fastest -> coalesced stores
    const int theta = pix0 >> 9;
    const int r0    = pix0 & 511;
    const int img0  = blockIdx.y * IMG_PER_GRP;

    // ---- per-pixel map (computed once, reused across IMG_PER_GRP images) ----
    // max_r = log(sqrt(H*H+W*W)/2 * 700)  in fp32, same op order as reference
    const float max_r = logf(sqrtf(524288.0f) * 0.5f * 700.0f);
    const float ang   = ((float)theta * 2.0f * 3.14159265358979323846f) / 512.0f;
    const float cs = cosf(ang);
    const float sn = sinf(ang);

    int   off[PPT][4];
    float wt [PPT][4];
    int   any = 0;

#pragma unroll
    for (int i = 0; i < PPT; ++i) {
        const float rf  = (float)(r0 + i);
        const float rad = expf((rf * max_r) / 512.0f);
        const float X   = 256.0f + rad * cs;   // cx + X0
        const float Y   = 256.0f - rad * sn;   // cy - Y0
        const bool inb  = (X >= 0.0f) & (X < 512.0f) & (Y >= 0.0f) & (Y < 512.0f);
        any |= (int)inb;

        int yd = (int)Y;  yd = yd < 0 ? 0 : (yd > 511 ? 511 : yd);  // trunc + clamp
        int xd = (int)X;  xd = xd < 0 ? 0 : (xd > 511 ? 511 : xd);
        const int yu = (yd + 1 > 511) ? 511 : yd + 1;
        const int xu = (xd + 1 > 511) ? 511 : xd + 1;

        float dyd = Y - (float)yd;  dyd *= dyd;
        float dyu = Y - (float)yu;  dyu *= dyu;
        float dxd = X - (float)xd;  dxd *= dxd;
        float dxu = X - (float)xu;  dxu *= dxu;

        const float dd = dyd + dxd, du = dyd + dxu;
        const float ud = dyu + dxd, uu = dyu + dxu;
        const float inv = inb ? (1.0f / (dd + du + ud + uu)) : 0.0f;  // mask folded in

        wt[i][0] = dd * inv;  wt[i][1] = du * inv;
        wt[i][2] = ud * inv;  wt[i][3] = uu * inv;
        off[i][0] = yd * 512 + xd;  off[i][1] = yd * 512 + xu;
        off[i][2] = yu * 512 + xd;  off[i][3] = yu * 512 + xu;
    }

    v4f* __restrict__ outv = (v4f*)(out + pix0);   // 16B aligned (pix0 % 4 == 0)

    // ---- fully masked: stream zeros (majority of the output, r >~ 242).
    // Non-temporal: output is write-once, keep it out of L2 so the gather
    // input (<=100MB) stays resident in the 192MB global L2.
    if (!any) {
        const v4f z = {0.0f, 0.0f, 0.0f, 0.0f};
        for (int img = img0; img < img0 + IMG_PER_GRP; ++img)
            __builtin_nontemporal_store(z, outv + img * (OUT_HW / 4));
        return;
    }

    // ---- gfx1250: warm L2 for the gather footprint (global_prefetch_b8) ----
    {
        const float* __restrict__ b0 = in + (size_t)img0 * OUT_HW;
#pragma unroll
        for (int i = 0; i < PPT; ++i)
#pragma unroll
            for (int k = 0; k < 4; ++k)
                __builtin_prefetch(b0 + off[i][k], 0, 3);
    }

    // ---- gather + weighted combine: 16 gathers + one 128b NT store per image ----
#pragma unroll 2
    for (int img = img0; img < img0 + IMG_PER_GRP; ++img) {
        const float* __restrict__ b = in + (size_t)img * OUT_HW;
        v4f v;
        v.x = wt[0][0]*b[off[0][0]] + wt[0][1]*b[off[0][1]]
            + wt[0][2]*b[off[0][2]] + wt[0][3]*b[off[0][3]];
        v.y = wt[1][0]*b[off[1][0]] + wt[1][1]*b[off[1][1]]
            + wt[1][2]*b[off[1][2]] + wt[1][3]*b[off[1][3]];
        v.z = wt[2][0]*b[off[2][0]] + wt[2][1]*b[off[2][1]]
            + wt[2][2]*b[off[2][2]] + wt[2][3]*b[off[2][3]];
        v.w = wt[3][0]*b[off[3][0]] + wt[3][1]*b[off[3][1]]
            + wt[3][2]*b[off[3][2]] + wt[3][3]*b[off[3][3]];
        __builtin_nontemporal_store(v, outv + img * (OUT_HW / 4));
    }
}

extern "C" void kernel_launch(void* const* d_in, const int* in_sizes, int n_in,
                              void* d_out, int out_size, void* d_ws, size_t ws_size,
                              hipStream_t stream)
{
    (void)in_sizes; (void)n_in; (void)out_size; (void)d_ws; (void)ws_size;
    const float* data = (const float*)d_in[0];
    float*       out  = (float*)d_out;

    // 262144 pixels / (256 threads * 4 pixels) = 256 blocks in x; 4 image groups in y
    dim3 grid(OUT_HW / (256 * PPT), IMG_GRP);
    dim3 block(256);
    logpolar_remap_kernel<<<grid, block, 0, stream>>>(data, out);
}